// PointerNet_19782619365575
// MI455X (gfx1250) — compile-verified
//
#include <hip/hip_runtime.h>
#include <hip/hip_bf16.h>
#include <math.h>

// Sizes from the reference
#define Bsz 128
#define Ssz 128
#define Esz 128
#define Hsz 128
#define TSTEPS 100
#define NEGV (-1000000000.0f)

typedef __attribute__((ext_vector_type(2))) float v2f;
typedef __attribute__((ext_vector_type(8))) float v8f;

// ---------------------------------------------------------------------------
// Kernel 1: tiny precompute  M2 = ctx2_w @ enc2_w  (128x3),
//           c2bias = ctx2_w @ enc2_b + ctx2_b      (128)
// ---------------------------------------------------------------------------
__global__ void prep_small(const float* __restrict__ ctx2_w,
                           const float* __restrict__ enc2_w,
                           const float* __restrict__ enc2_b,
                           const float* __restrict__ ctx2_b,
                           float* __restrict__ M2, float* __restrict__ c2bias) {
  int h = threadIdx.x;
  if (h >= Hsz) return;
  float m0 = 0.f, m1 = 0.f, m2 = 0.f, cb = 0.f;
  for (int e = 0; e < Esz; e++) {
    float w = ctx2_w[h * Esz + e];
    m0 += w * enc2_w[e * 3 + 0];
    m1 += w * enc2_w[e * 3 + 1];
    m2 += w * enc2_w[e * 3 + 2];
    cb += w * enc2_b[e];
  }
  M2[h * 3 + 0] = m0; M2[h * 3 + 1] = m1; M2[h * 3 + 2] = m2;
  c2bias[h] = cb + ctx2_b[h];
}

// ---------------------------------------------------------------------------
// Kernel 2: emb_static[b][e][s] = static[b,0,s]*enc1_w[e,0]
//                               + static[b,1,s]*enc1_w[e,1] + enc1_b[e]
// ---------------------------------------------------------------------------
__global__ void emb_static_k(const float* __restrict__ stat,
                             const float* __restrict__ enc1_w,
                             const float* __restrict__ enc1_b,
                             float* __restrict__ emb) {
  int n = blockIdx.x * blockDim.x + threadIdx.x;
  if (n >= Bsz * Esz * Ssz) return;
  int s = n & 127;
  int e = (n >> 7) & 127;
  int b = n >> 14;
  float v = stat[(size_t)b * 2 * Ssz + s] * enc1_w[e * 2 + 0] +
            stat[(size_t)b * 2 * Ssz + Ssz + s] * enc1_w[e * 2 + 1] + enc1_b[e];
  emb[n] = v;
}

// ---------------------------------------------------------------------------
// Kernel 3: WMMA f32 batched GEMM  base[b][s][h] = (ctx1_w @ emb[b])[h][s]
//           + ctx1_b[h] + c2bias[h] + M2[h,0]*dyn[b,0,s] + M2[h,1]*dyn[b,1,s]
// One wave32 per 16x16 tile; K=128 in steps of 4 via V_WMMA_F32_16X16X4_F32.
// A layout (ISA 7.12.2, 32-bit A 16x4): lanes 0-15 -> K=k0,k0+1 ;
// lanes 16-31 -> K=k0+2,k0+3 ; M = lane%16.  B layout mirrored per row.
// C/D: VGPR v -> M = v (lanes<16) / v+8 (lanes>=16), N = lane%16.
// ---------------------------------------------------------------------------
__global__ __launch_bounds__(256) void estatic_wmma(
    const float* __restrict__ A,        // ctx1_w, HxE row-major
    const float* __restrict__ emb,      // [b][e][s]
    const float* __restrict__ ctx1_b,
    const float* __restrict__ c2bias,
    const float* __restrict__ M2,
    const float* __restrict__ dynamic,  // [b][3][s]
    float* __restrict__ base) {         // [b][s][h]
  int wave = threadIdx.x >> 5;
  int lane = threadIdx.x & 31;
  int b = blockIdx.x >> 3;
  int tile = ((blockIdx.x & 7) << 3) + wave;  // 0..63
  int mh = (tile >> 3) << 4;                  // h tile base
  int ns = (tile & 7) << 4;                   // s tile base
  const float* Bm = emb + (size_t)b * Esz * Ssz;
  int lhalf = (lane < 16) ? 0 : 2;
  int lm = lane & 15;
  v8f acc = {};
  for (int k0 = 0; k0 < Esz; k0 += 4) {
    v2f a, bf;
    a.x = A[(mh + lm) * Esz + k0 + lhalf];
    a.y = A[(mh + lm) * Esz + k0 + lhalf + 1];
    bf.x = Bm[(size_t)(k0 + lhalf) * Ssz + ns + lm];
    bf.y = Bm[(size_t)(k0 + lhalf + 1) * Ssz + ns + lm];
    acc = __builtin_amdgcn_wmma_f32_16x16x4_f32(
        /*neg_a=*/false, a, /*neg_b=*/false, bf,
        /*c_mod=*/(short)0, acc, /*reuse_a=*/false, /*reuse_b=*/false);
  }
  // Fused epilogue: each lane owns 8 contiguous h at column s.
  int s = ns + lm;
  int h0 = mh + ((lane < 16) ? 0 : 8);
  float d0 = dynamic[(size_t)b * 3 * Ssz + s];
  float d1 = dynamic[(size_t)b * 3 * Ssz + Ssz + s];
  float* outp = base + ((size_t)(b * Ssz + s)) * Hsz + h0;
#pragma unroll
  for (int v = 0; v < 8; v++) {
    int h = h0 + v;
    outp[v] = acc[v] + ctx1_b[h] + c2bias[h] + M2[h * 3 + 0] * d0 + M2[h * 3 + 1] * d1;
  }
}

// ---------------------------------------------------------------------------
// Kernel 4: persistent per-batch 100-step pointer-net scan.
// One block (8 wave32, 256 threads) per batch; all carry state in LDS.
// ---------------------------------------------------------------------------
__global__ __launch_bounds__(256) void pointer_steps(
    const float* __restrict__ base,      // [b][s][h]
    const float* __restrict__ emb,       // [b][e][s]
    const float* __restrict__ M2,        // [h][3]
    const float* __restrict__ distances, // [b][s][s]
    const float* __restrict__ dynamic,   // [b][3][s]
    const float* __restrict__ V0, const float* __restrict__ V1,
    const float* __restrict__ V2,
    const float* __restrict__ W_ih, const float* __restrict__ W_hh,
    const float* __restrict__ b_ih, const float* __restrict__ b_hh,
    const float* __restrict__ att_w, const float* __restrict__ att_b,
    const float* __restrict__ lin_w, const float* __restrict__ lin_b,
    float* __restrict__ out) {
  __shared__ float sh_h[Hsz], sh_c[Hsz], sh_ht[Hsz], sh_mask[Ssz], sh_dyn2[Ssz],
      sh_decin[Esz], sh_eplus[Hsz], sh_alpha[Ssz], sh_m[Ssz], sh_ctx[Esz],
      sh_part[512];
  __shared__ int red_i[128];
  __shared__ int sh_idx;
  __shared__ float sh_plog;

  int tid = threadIdx.x;
  int b = blockIdx.x;
  const float* embB = emb + (size_t)b * Esz * Ssz;
  const float* baseB = base + (size_t)b * Ssz * Hsz;

  // ---- init carry ----
  if (tid < Hsz) { sh_h[tid] = 0.f; sh_c[tid] = 0.f; }
  if (tid == 0) sh_plog = 0.f;
  if (tid < Ssz) {
    int s = tid;
    const float* db = distances + (size_t)b * Ssz * Ssz;
    float dba;
    if (s == 0) {
      dba = 0.f;  // dis_by_afs.at[:,0].set(0.0)
    } else {
      dba = 1e30f;
      for (int a = 1; a <= 3; a++)
        dba = fminf(dba, db[a * Ssz] + db[a * Ssz + s]);
    }
    bool cond = (db[s] > 150.0f) && (dba > 300.0f);  // dis_max=300
    float m = (s == 0) ? NEGV : 0.f;
    if (cond) m = NEGV;
    sh_mask[s] = m;
    float dy = dynamic[(size_t)b * 3 * Ssz + 2 * Ssz + s];
    sh_dyn2[s] = cond ? 0.f : dy;
    sh_decin[s] = embB[(size_t)s * Ssz];  // emb_static[b, e=s, 0]
  }
  __syncthreads();

  for (int t = 0; t < TSTEPS; t++) {
    // ---- LSTM gates: 512-wide matvec, 2 rows/thread ----
    for (int rep = 0; rep < 2; rep++) {
      int j = tid + rep * 256;
      float acc = b_ih[j] + b_hh[j];
      const float4* wi = (const float4*)(W_ih + (size_t)j * Esz);
      const float4* wh = (const float4*)(W_hh + (size_t)j * Hsz);
#pragma unroll 4
      for (int e = 0; e < Esz / 4; e++) {
        float4 w = wi[e];
        acc += w.x * sh_decin[e * 4] + w.y * sh_decin[e * 4 + 1] +
               w.z * sh_decin[e * 4 + 2] + w.w * sh_decin[e * 4 + 3];
      }
#pragma unroll 4
      for (int h = 0; h < Hsz / 4; h++) {
        float4 w = wh[h];
        acc += w.x * sh_h[h * 4] + w.y * sh_h[h * 4 + 1] +
               w.z * sh_h[h * 4 + 2] + w.w * sh_h[h * 4 + 3];
      }
      sh_part[j] = acc;  // gates live in sh_part[0..511]
    }
    __syncthreads();
    // ---- LSTM cell ----
    if (tid < Hsz) {
      int h = tid;
      float gi = sh_part[h], gf = sh_part[Hsz + h];
      float gg = sh_part[2 * Hsz + h], go = sh_part[3 * Hsz + h];
      float si = 1.f / (1.f + __expf(-gi));
      float sf = 1.f / (1.f + __expf(-gf));
      float so = 1.f / (1.f + __expf(-go));
      float c2 = sf * sh_c[h] + si * tanhf(gg);
      sh_c[h] = c2;
      sh_ht[h] = so * tanhf(c2);
    }
    __syncthreads();
    // ---- e_plus = att_w @ h_t + att_b ----
    if (tid < Hsz) {
      int h = tid;
      float acc = att_b[h];
      const float4* aw = (const float4*)(att_w + (size_t)h * Esz);
#pragma unroll 4
      for (int e = 0; e < Esz / 4; e++) {
        float4 w = aw[e];
        acc += w.x * sh_ht[e * 4] + w.y * sh_ht[e * 4 + 1] +
               w.z * sh_ht[e * 4 + 2] + w.w * sh_ht[e * 4 + 3];
      }
      sh_eplus[h] = acc;
    }
    __syncthreads();
    // ---- logits[s] = sum_h Vsel(s)[h]*tanh(base + M2[h,2]*dyn2[s] + eplus[h])
    {
      int s = tid >> 1, half = tid & 1;
      const float* Vp = (s == 0) ? V0 : ((s <= 3) ? V1 : V2);
      const float* bp = baseB + (size_t)s * Hsz;
      float d2 = sh_dyn2[s];
      float acc = 0.f;
      int h0 = half * 64;
      for (int h = h0; h < h0 + 64; h++) {
        float tv = tanhf(bp[h] + M2[h * 3 + 2] * d2 + sh_eplus[h]);
        acc += Vp[h] * tv;
      }
      sh_part[tid] = acc;
    }
    __syncthreads();
    if (tid < Ssz) sh_m[tid] = sh_part[2 * tid] + sh_part[2 * tid + 1] + sh_mask[tid];
    __syncthreads();
    // ---- argmax over s (first-index tie-break, matches jnp.argmax) ----
    if (tid < 128) { sh_part[tid] = sh_m[tid]; red_i[tid] = tid; }
    __syncthreads();
    for (int off = 64; off > 0; off >>= 1) {
      if (tid < off) {
        float vv = sh_part[tid + off];
        int ii = red_i[tid + off];
        if (vv > sh_part[tid] || (vv == sh_part[tid] && ii < red_i[tid])) {
          sh_part[tid] = vv;
          red_i[tid] = ii;
        }
      }
      __syncthreads();
    }
    float maxv = sh_part[0];
    if (tid == 0) sh_idx = red_i[0];
    __syncthreads();
    // ---- softmax ----
    if (tid < Ssz) {
      float e = __expf(sh_m[tid] - maxv);
      sh_alpha[tid] = e;
      sh_part[tid] = e;
    }
    __syncthreads();
    for (int off = 64; off > 0; off >>= 1) {
      if (tid < off) sh_part[tid] += sh_part[tid + off];
      __syncthreads();
    }
    float sumexp = sh_part[0];
    if (tid < Ssz) sh_alpha[tid] = sh_alpha[tid] / sumexp;
    __syncthreads();
    int idx = sh_idx;
    if (tid == 0) {
      sh_plog += __logf(sh_alpha[idx]);
      out[(size_t)b * TSTEPS + t] = (float)idx;  // tour index
      sh_dyn2[idx] = 0.f;
      if (idx > 3) sh_mask[idx] = NEGV;  // customers masked after visit
    }
    __syncthreads();
    // ---- ctx[e] = sum_s emb[b,e,s]*alpha[s] ----
    {
      int e = tid >> 1, half = tid & 1;
      const float* ep = embB + (size_t)e * Ssz + half * 64;
      const float* ap = sh_alpha + half * 64;
      float acc = 0.f;
      for (int s = 0; s < 64; s++) acc += ep[s] * ap[s];
      sh_part[tid] = acc;
    }
    __syncthreads();
    if (tid < Esz) sh_ctx[tid] = sh_part[2 * tid] + sh_part[2 * tid + 1];
    __syncthreads();
    // ---- h_new = lin_w @ [h_t; ctx] + lin_b ;  dec_in = emb[:, idx] ----
    if (tid < Hsz) {
      int h = tid;
      float acc = lin_b[h];
      const float4* lw = (const float4*)(lin_w + (size_t)h * (Esz + Hsz));
#pragma unroll 4
      for (int k = 0; k < Hsz / 4; k++) {
        float4 w = lw[k];
        acc += w.x * sh_ht[k * 4] + w.y * sh_ht[k * 4 + 1] +
               w.z * sh_ht[k * 4 + 2] + w.w * sh_ht[k * 4 + 3];
      }
#pragma unroll 4
      for (int k = 0; k < Esz / 4; k++) {
        float4 w = lw[Hsz / 4 + k];
        acc += w.x * sh_ctx[k * 4] + w.y * sh_ctx[k * 4 + 1] +
               w.z * sh_ctx[k * 4 + 2] + w.w * sh_ctx[k * 4 + 3];
      }
      sh_h[h] = acc;
      sh_decin[h] = embB[(size_t)h * Ssz + idx];
    }
    __syncthreads();
  }
  if (tid == 0) out[(size_t)Bsz * TSTEPS + b] = sh_plog;
}

// ---------------------------------------------------------------------------
extern "C" void kernel_launch(void* const* d_in, const int* in_sizes, int n_in,
                              void* d_out, int out_size, void* d_ws,
                              size_t ws_size, hipStream_t stream) {
  (void)in_sizes; (void)n_in; (void)out_size; (void)ws_size;
  const float* stat      = (const float*)d_in[0];
  const float* dynamic_  = (const float*)d_in[1];
  const float* distances = (const float*)d_in[2];
  const float* enc1_w = (const float*)d_in[3];
  const float* enc1_b = (const float*)d_in[4];
  const float* enc2_w = (const float*)d_in[5];
  const float* enc2_b = (const float*)d_in[6];
  const float* ctx1_w = (const float*)d_in[7];
  const float* ctx1_b = (const float*)d_in[8];
  const float* ctx2_w = (const float*)d_in[9];
  const float* ctx2_b = (const float*)d_in[10];
  const float* att_w  = (const float*)d_in[11];
  const float* att_b  = (const float*)d_in[12];
  const float* V0 = (const float*)d_in[13];
  const float* V1 = (const float*)d_in[14];
  const float* V2 = (const float*)d_in[15];
  const float* W_ih = (const float*)d_in[16];
  const float* W_hh = (const float*)d_in[17];
  const float* b_ih = (const float*)d_in[18];
  const float* b_hh = (const float*)d_in[19];
  const float* lin_w = (const float*)d_in[20];
  const float* lin_b = (const float*)d_in[21];

  // Workspace: M2 (384) | c2bias (128) | emb_static (2M) | base (2M) = ~16.8 MB
  float* ws = (float*)d_ws;
  float* M2 = ws;
  float* c2bias = ws + 384;
  float* emb = ws + 512;
  float* base = emb + (size_t)Bsz * Esz * Ssz;
  float* out = (float*)d_out;

  prep_small<<<1, 128, 0, stream>>>(ctx2_w, enc2_w, enc2_b, ctx2_b, M2, c2bias);
  emb_static_k<<<(Bsz * Esz * Ssz) / 256, 256, 0, stream>>>(stat, enc1_w, enc1_b, emb);
  estatic_wmma<<<Bsz * 8, 256, 0, stream>>>(ctx1_w, emb, ctx1_b, c2bias, M2,
                                            dynamic_, base);
  pointer_steps<<<Bsz, 256, 0, stream>>>(base, emb, M2, distances, dynamic_,
                                         V0, V1, V2, W_ih, W_hh, b_ih, b_hh,
                                         att_w, att_b, lin_w, lin_b, out);
}